// MultiUTAETemporalExtractor_11501922419394
// MI455X (gfx1250) — compile-verified
//
#include <hip/hip_runtime.h>

// ---- problem geometry (from reference) ----
#define B_          16
#define T_          24
#define C_          128
#define ROW_LEN     1024            // Hf*Wf = 32*32
#define ROWS_TOTAL  (B_*T_*C_)      // 49152 mean-reduction rows
#define PIX_PER_BT  16384           // H*W = 128*128
#define NCLS        6

// output layout (floats): [emb 49152][labels 384][sits 16][positions 384]
#define OFF_EMB   0
#define OFF_LAB   49152
#define OFF_SITS  49536
#define OFF_POS   49552

typedef __attribute__((ext_vector_type(2))) float v2f;
typedef __attribute__((ext_vector_type(4))) float v4f;
typedef __attribute__((ext_vector_type(8))) float v8f;
typedef __attribute__((ext_vector_type(4))) int   v4i;

// ---------------------------------------------------------------------------
// Kernel 1: spatial mean. One wave32 per contiguous 1024-float row.
// 8 NT b128 loads per lane (coalesced), lane-partial sum, then a single
// v_wmma_f32_16x16x4_f32 against B=ones folds lane L and L+16 into row L%16;
// summing the 8 accumulator VGPRs + two shuffles yields the full 32-lane sum.
// ---------------------------------------------------------------------------
__global__ __launch_bounds__(256) void emb_mean_kernel(const float* __restrict__ x,
                                                       float* __restrict__ out) {
  const int lane = threadIdx.x & 31;
  const int wave = threadIdx.x >> 5;
  const int row  = blockIdx.x * 8 + wave;           // 6144 blocks * 8 waves = 49152
  const float* p = x + (size_t)row * ROW_LEN;

  float acc = 0.0f;
#pragma unroll
  for (int i = 0; i < 8; ++i) {
    const v4f* q = (const v4f*)(p + i * 128 + lane * 4);   // 16B-aligned, coalesced
    v4f v = __builtin_nontemporal_load(q);                 // read-once stream: NT
    acc += (v.x + v.y) + (v.z + v.w);
  }

  // Cross-lane reduction via WMMA: D[m][n] = acc(lane m) + acc(lane m+16).
  v2f a; a.x = acc;  a.y = 0.0f;     // lane<16 -> (M=lane,K=0/1); lane>=16 -> (M=lane-16,K=2/3)
  v2f b; b.x = 1.0f; b.y = 1.0f;     // B = all-ones 4x16
  v8f c = {};
  c = __builtin_amdgcn_wmma_f32_16x16x4_f32(false, a, false, b, (short)0, c, false, false);

  // lanes 0-15 hold rows 0..7 in c[0..7]; lanes 16-31 hold rows 8..15.
  float half = ((c[0] + c[1]) + (c[2] + c[3])) + ((c[4] + c[5]) + (c[6] + c[7]));
  float total = __shfl(half, 0, 32) + __shfl(half, 16, 32);

  if (lane == 0) out[OFF_EMB + row] = total * (1.0f / (float)ROW_LEN);
}

// ---------------------------------------------------------------------------
// Kernel 2: per-(B,T) majority class over 16384 labels. Block per (b,t).
// int4 NT loads, register counters, 6 LDS atomics/thread, first-max argmax.
// ---------------------------------------------------------------------------
__global__ __launch_bounds__(256) void labels_kernel(const int* __restrict__ gt,
                                                     float* __restrict__ out) {
  __shared__ int cnt[NCLS];
  if (threadIdx.x < NCLS) cnt[threadIdx.x] = 0;
  __syncthreads();

  const int bt = blockIdx.x;                                  // 0..383
  const v4i* p = (const v4i*)(gt + (size_t)bt * PIX_PER_BT);  // 16B-aligned
  int c0 = 0, c1 = 0, c2 = 0, c3 = 0, c4 = 0, c5 = 0;

#pragma unroll 4
  for (int i = threadIdx.x; i < PIX_PER_BT / 4; i += 256) {
    v4i v = __builtin_nontemporal_load(p + i);
    int vv[4] = { v.x, v.y, v.z, v.w };
#pragma unroll
    for (int j = 0; j < 4; ++j) {
      c0 += (vv[j] == 0); c1 += (vv[j] == 1); c2 += (vv[j] == 2);
      c3 += (vv[j] == 3); c4 += (vv[j] == 4); c5 += (vv[j] == 5);
    }
  }
  atomicAdd(&cnt[0], c0); atomicAdd(&cnt[1], c1); atomicAdd(&cnt[2], c2);
  atomicAdd(&cnt[3], c3); atomicAdd(&cnt[4], c4); atomicAdd(&cnt[5], c5);
  __syncthreads();

  if (threadIdx.x == 0) {
    int best = 0, bc = cnt[0];
#pragma unroll
    for (int k = 1; k < NCLS; ++k) {          // strict '>' => first max (jnp.argmax)
      if (cnt[k] > bc) { bc = cnt[k]; best = k; }
    }
    out[OFF_LAB + bt] = (float)best;
  }
}

// ---------------------------------------------------------------------------
// Kernel 3: pass-through of sits_id (16) and positions (384) as floats.
// ---------------------------------------------------------------------------
__global__ void tail_copy_kernel(const int* __restrict__ positions,
                                 const int* __restrict__ sits_id,
                                 float* __restrict__ out) {
  int i = blockIdx.x * blockDim.x + threadIdx.x;
  if (i < 16)  out[OFF_SITS + i] = (float)sits_id[i];
  if (i < 384) out[OFF_POS + i]  = (float)positions[i];
}

// ---------------------------------------------------------------------------
extern "C" void kernel_launch(void* const* d_in, const int* in_sizes, int n_in,
                              void* d_out, int out_size, void* d_ws, size_t ws_size,
                              hipStream_t stream) {
  const float* x         = (const float*)d_in[0];  // out_temporal (B,T,C,32,32) f32
  const int*   gt        = (const int*)  d_in[1];  // gt (B,T,128,128)
  const int*   positions = (const int*)  d_in[2];  // (B,T)
  const int*   sits_id   = (const int*)  d_in[3];  // (B,)
  float* out = (float*)d_out;

  emb_mean_kernel<<<ROWS_TOTAL / 8, 256, 0, stream>>>(x, out);
  labels_kernel<<<B_ * T_, 256, 0, stream>>>(gt, out);
  tail_copy_kernel<<<2, 256, 0, stream>>>(positions, sits_id, out);
}